// MlpSeqAttention_32409823216291
// MI455X (gfx1250) — compile-verified
//
#include <hip/hip_runtime.h>
#include <hip/hip_bf16.h>

typedef __attribute__((ext_vector_type(2))) float v2f;
typedef __attribute__((ext_vector_type(8))) float v8f;

#define B_ 4
#define Q_ 256
#define K_ 1024
#define D_ 256   // DQ == DK == 256

// tanh(x) = 1 - 2/(1+exp(2x)); exp via v_exp_f32, divide via v_rcp_f32.
// Saturates correctly: x>>0 -> exp2->inf -> rcp->0 -> 1;  x<<0 -> exp2->0 -> 1-2 = -1.
__device__ __forceinline__ float tanh_fast(float x) {
    float e = __builtin_amdgcn_exp2f(x * 2.88539008177793f);  // 2*log2(e)
    return 1.0f - 2.0f * __builtin_amdgcn_rcpf(e + 1.0f);
}

// -------------------------------------------------------------------------
// Y[M x 256] = X[M x 256] @ W[256 x 256]  (row-major f32) via v_wmma_f32_16x16x4_f32
// block: 256 threads = 8 waves. Block tile: 16 rows x 128 cols (one 16x16 tile per wave).
// grid: (256/128 = 2, M/16)
// -------------------------------------------------------------------------
__global__ __launch_bounds__(256)
void proj_gemm_wmma(const float* __restrict__ X, const float* __restrict__ W,
                    float* __restrict__ Y) {
    __shared__ float xs[16][D_ + 4];   // pad: row stride 260 -> conflict-free A reads
    const int tid = threadIdx.x;
    const int m0  = blockIdx.y * 16;
    const int n0  = blockIdx.x * 128;

    #pragma unroll
    for (int i = 0; i < 16; ++i)                    // coalesced stage of X tile
        xs[i][tid] = X[(size_t)(m0 + i) * D_ + tid];
    __syncthreads();

    const int lane = tid & 31;
    const int wave = tid >> 5;
    const int mn   = lane & 15;     // A row / B col within tile
    const int hi   = lane >> 4;     // 0: K pair {0,1}; 1: K pair {2,3}
    const int nw   = n0 + wave * 16;

    v8f c = {};
    for (int kk = 0; kk < D_; kk += 4) {
        v2f a, b;
        a.x = xs[mn][kk + 2 * hi + 0];
        a.y = xs[mn][kk + 2 * hi + 1];
        b.x = W[(size_t)(kk + 2 * hi + 0) * D_ + nw + mn];
        b.y = W[(size_t)(kk + 2 * hi + 1) * D_ + nw + mn];
        c = __builtin_amdgcn_wmma_f32_16x16x4_f32(false, a, false, b,
                                                  (short)0, c, false, false);
    }
    // D layout: VGPR r -> row (r + 8*hi), col mn
    #pragma unroll
    for (int r = 0; r < 8; ++r)
        Y[(size_t)(m0 + r + 8 * hi) * D_ + nw + mn] = c[r];
}

// -------------------------------------------------------------------------
// scores[b,q,k] = mask[b,k] ? sum_e tanh(qp[b,q,e]+kp[b,k,e])*wv[e] : -inf
// block: 256 threads = one 16q x 16k tile; grid (K/16, Q/16, B). TRANS-bound hot loop.
// -------------------------------------------------------------------------
__global__ __launch_bounds__(256)
void scores_kernel(const float* __restrict__ qp, const float* __restrict__ kp,
                   const float* __restrict__ wv, const int* __restrict__ mask,
                   float* __restrict__ scores) {
    __shared__ float qs[16][D_ + 4];
    __shared__ float ks[16][D_ + 4];
    __shared__ float wvs[D_];
    const int tid = threadIdx.x;
    const int k0 = blockIdx.x * 16, q0 = blockIdx.y * 16, b = blockIdx.z;

    const float* qpb = qp + ((size_t)b * Q_ + q0) * D_;
    const float* kpb = kp + ((size_t)b * K_ + k0) * D_;
    #pragma unroll
    for (int i = 0; i < 16; ++i) {
        qs[i][tid] = qpb[(size_t)i * D_ + tid];
        ks[i][tid] = kpb[(size_t)i * D_ + tid];
    }
    wvs[tid] = wv[tid];
    __syncthreads();

    const int tk = tid & 15, tq = tid >> 4;
    float s = 0.f;
    #pragma unroll 4
    for (int e = 0; e < D_; ++e)
        s += tanh_fast(qs[tq][e] + ks[tk][e]) * wvs[e];

    const int kg = k0 + tk;
    const bool valid = mask[(size_t)b * K_ + kg] != 0;
    scores[((size_t)b * Q_ + q0 + tq) * K_ + kg] = valid ? s : -__builtin_inff();
}

// -------------------------------------------------------------------------
// In-place softmax over K=1024 per (b,q) row. grid: B*Q blocks, 256 threads.
// -------------------------------------------------------------------------
__global__ __launch_bounds__(256)
void softmax_kernel(float* __restrict__ scores) {
    __shared__ float red[256];
    float* s = scores + (size_t)blockIdx.x * K_;
    const int tid = threadIdx.x;

    float v0[4];
    float mx = -__builtin_inff();
    #pragma unroll
    for (int i = 0; i < 4; ++i) { v0[i] = s[tid + i * 256]; mx = fmaxf(mx, v0[i]); }
    red[tid] = mx; __syncthreads();
    for (int off = 128; off > 0; off >>= 1) {
        if (tid < off) red[tid] = fmaxf(red[tid], red[tid + off]);
        __syncthreads();
    }
    mx = red[0]; __syncthreads();

    float ex[4], sum = 0.f;
    #pragma unroll
    for (int i = 0; i < 4; ++i) {
        ex[i] = __builtin_amdgcn_exp2f((v0[i] - mx) * 1.44269504088896f);
        sum += ex[i];
    }
    red[tid] = sum; __syncthreads();
    for (int off = 128; off > 0; off >>= 1) {
        if (tid < off) red[tid] += red[tid + off];
        __syncthreads();
    }
    const float inv = 1.0f / red[0];
    #pragma unroll
    for (int i = 0; i < 4; ++i) s[tid + i * 256] = ex[i] * inv;
}

// -------------------------------------------------------------------------
// out[b,q,d] = sum_k probs[b,q,k] * v[b,k,d] via v_wmma_f32_16x16x4_f32.
// block: 256 threads = 8 waves; block tile = 16q x 256d (each wave: two 16-wide d tiles,
// sharing one A load per WMMA pair). grid: (Q/16, B). probs tile staged in LDS (~66KB).
// -------------------------------------------------------------------------
__global__ __launch_bounds__(256)
void attend_wmma(const float* __restrict__ probs, const float* __restrict__ V,
                 float* __restrict__ out) {
    __shared__ float ps[16][K_ + 4];
    const int tid = threadIdx.x;
    const int q0 = blockIdx.x * 16, b = blockIdx.y;

    const float* pb = probs + ((size_t)b * Q_ + q0) * K_;
    #pragma unroll
    for (int i = 0; i < 64; ++i) {                 // 16*1024 elems / 256 threads
        int f = i * 256 + tid;
        ps[f >> 10][f & 1023] = pb[(size_t)(f >> 10) * K_ + (f & 1023)];
    }
    __syncthreads();

    const int lane = tid & 31, wave = tid >> 5;
    const int mn = lane & 15, hi = lane >> 4;
    const int d0 = wave * 32;
    const float* vb = V + (size_t)b * K_ * D_;

    v8f c0 = {}, c1 = {};
    for (int kk = 0; kk < K_; kk += 4) {
        v2f a, b0, b1;
        a.x = ps[mn][kk + 2 * hi + 0];
        a.y = ps[mn][kk + 2 * hi + 1];
        const float* vr0 = vb + (size_t)(kk + 2 * hi + 0) * D_;
        const float* vr1 = vb + (size_t)(kk + 2 * hi + 1) * D_;
        b0.x = vr0[d0 + mn];       b0.y = vr1[d0 + mn];
        b1.x = vr0[d0 + 16 + mn];  b1.y = vr1[d0 + 16 + mn];
        c0 = __builtin_amdgcn_wmma_f32_16x16x4_f32(false, a, false, b0,
                                                   (short)0, c0, false, false);
        c1 = __builtin_amdgcn_wmma_f32_16x16x4_f32(false, a, false, b1,
                                                   (short)0, c1, false, false);
    }
    float* ob = out + ((size_t)b * Q_ + q0) * D_;
    #pragma unroll
    for (int r = 0; r < 8; ++r) {
        ob[(size_t)(r + 8 * hi) * D_ + d0 + mn]      = c0[r];
        ob[(size_t)(r + 8 * hi) * D_ + d0 + 16 + mn] = c1[r];
    }
}

extern "C" void kernel_launch(void* const* d_in, const int* in_sizes, int n_in,
                              void* d_out, int out_size, void* d_ws, size_t ws_size,
                              hipStream_t stream) {
    const float* q    = (const float*)d_in[0];   // [B,Q,DQ]
    const float* k    = (const float*)d_in[1];   // [B,K,DK]
    const float* v    = (const float*)d_in[2];   // [B,K,DK]
    const int*   mask = (const int*)  d_in[3];   // [B,K]
    const float* Wq   = (const float*)d_in[4];   // [DQ,DK]
    const float* Wk   = (const float*)d_in[5];   // [DK,DK]
    const float* wv   = (const float*)d_in[6];   // [DK]
    float* out = (float*)d_out;                  // [B,Q,DK]

    float* qp = (float*)d_ws;                    // B*Q*D  = 1 MB
    float* kp = qp + (size_t)B_ * Q_ * D_;       // B*K*D  = 4 MB
    float* sc = kp + (size_t)B_ * K_ * D_;       // B*Q*K  = 4 MB (scores -> probs in place)

    proj_gemm_wmma<<<dim3(2, (B_ * Q_) / 16), 256, 0, stream>>>(q, Wq, qp);
    proj_gemm_wmma<<<dim3(2, (B_ * K_) / 16), 256, 0, stream>>>(k, Wk, kp);
    scores_kernel<<<dim3(K_ / 16, Q_ / 16, B_), 256, 0, stream>>>(qp, kp, wv, mask, sc);
    softmax_kernel<<<B_ * Q_, 256, 0, stream>>>(sc);
    attend_wmma<<<dim3(Q_ / 16, B_), 256, 0, stream>>>(sc, v, out);
}